// AFResampler_78675210928223
// MI455X (gfx1250) — compile-verified
//
#include <hip/hip_runtime.h>
#include <stdint.h>

typedef __attribute__((ext_vector_type(2))) float        v2f;
typedef __attribute__((ext_vector_type(8))) float        v8f;
typedef __attribute__((ext_vector_type(4))) unsigned int v4u;
typedef __attribute__((ext_vector_type(8))) int          v8i;
typedef __attribute__((ext_vector_type(4))) int          v4i;

#define B_     8
#define C_     64
#define H_     256
#define W_     256
#define OH_    128
#define OW_    128
#define KTOT   576      // 64 in-channels * 9 taps
#define KPAIRS 288      // KTOT / 2

// ---------------------------------------------------------------------------
// Tiny MLP: 4 -> 256 -> 256 -> 1, evaluated for the 9 (ow,oh) offsets.
// Output: wout[0..8] = per-offset scalar weights, wout[9] = their sum.
// ---------------------------------------------------------------------------
__global__ void mlp_weights_kernel(const float* __restrict__ w1, const float* __restrict__ b1,
                                   const float* __restrict__ w2, const float* __restrict__ b2,
                                   const float* __restrict__ w3, const float* __restrict__ b3,
                                   float* __restrict__ wout) {
  __shared__ float h1s[256];
  __shared__ float red[256];
  const int j = threadIdx.x;
  float wsum = 0.0f;
  int idx = 0;
  for (int oh = -1; oh <= 1; ++oh) {
    for (int ow = -1; ow <= 1; ++ow, ++idx) {
      // layer 1 (sH = sW = 2.0 for 256 -> 128)
      float h = b1[j] + (float)ow * w1[j] + (float)oh * w1[256 + j]
              + 2.0f * w1[512 + j] + 2.0f * w1[768 + j];
      h1s[j] = fmaxf(h, 0.0f);
      __syncthreads();
      // layer 2
      float h2 = b2[j];
      for (int k = 0; k < 256; ++k) h2 = fmaf(h1s[k], w2[k * 256 + j], h2);
      h2 = fmaxf(h2, 0.0f);
      // layer 3 dot + tree reduction
      red[j] = h2 * w3[j];
      __syncthreads();
      for (int s = 128; s > 0; s >>= 1) {
        if (j < s) red[j] += red[j + s];
        __syncthreads();
      }
      if (j == 0) {
        float wv = red[0] + b3[0];
        wout[idx] = wv;
        wsum += wv;
      }
      __syncthreads();  // protect h1s/red before next offset
    }
  }
  if (j == 0) wout[9] = wsum;
}

// ---------------------------------------------------------------------------
// Pack conv weights OIHW -> [k/2][n][2] pairs with k = (ky*3+kx)*64 + c,
// so each WMMA lane fetches its two B elements as one 8-byte LDS load.
// conv2 is zero-padded from 3 to 16 output channels.
// ---------------------------------------------------------------------------
__global__ void pack_weights_kernel(const float* __restrict__ conv1_w,
                                    const float* __restrict__ conv2_w,
                                    float* __restrict__ bp1, float* __restrict__ bp2) {
  const int t = blockIdx.x * blockDim.x + threadIdx.x;
  if (t < KPAIRS * 128) {                 // conv1: [288][64][2]
    const int kk = t >> 7, r = t & 127;
    const int n = r >> 1, jj = r & 1;
    const int k = kk * 2 + jj;
    const int c = k & 63, s = k >> 6;     // s = ky*3+kx
    bp1[t] = conv1_w[(n * 64 + c) * 9 + s];
  } else {
    const int d = t - KPAIRS * 128;
    if (d < KPAIRS * 32) {                // conv2 padded: [288][16][2]
      const int kk = d >> 5, r = d & 31;
      const int n = r >> 1, jj = r & 1;
      const int k = kk * 2 + jj;
      const int c = k & 63, s = k >> 6;
      bp2[d] = (n < 3) ? conv2_w[(n * 64 + c) * 9 + s] : 0.0f;
    }
  }
}

// ---------------------------------------------------------------------------
// Bilinear resample (align_corners chain, identical for all 9 offsets since
// normalize() is shift-invariant) + faithful 9-term weighting / w_sum.
// Output in NHWC [8][128][128][64] for the implicit-GEMM conv that follows.
// ---------------------------------------------------------------------------
__global__ void resample_kernel(const float* __restrict__ feat,
                                const float* __restrict__ wv,
                                float* __restrict__ mid) {
  const int t  = blockIdx.x * blockDim.x + threadIdx.x;
  const int ox = t & (OW_ - 1);
  const int oy = (t >> 7) & (OH_ - 1);
  const int c  = (t >> 14) & (C_ - 1);
  const int b  = t >> 20;

  const float cx = ((float)ox + 0.5f) * 2.0f - 0.5f;
  const float cy = ((float)oy + 0.5f) * 2.0f - 0.5f;
  const float gx = (cx - 0.5f) * (2.0f / 254.0f) - 1.0f;
  const float gy = (cy - 0.5f) * (2.0f / 254.0f) - 1.0f;
  const float ix = (gx + 1.0f) * 0.5f * (float)(W_ - 1);
  const float iy = (gy + 1.0f) * 0.5f * (float)(H_ - 1);
  const float fx0 = floorf(ix), fy0 = floorf(iy);
  const int x0 = (int)fx0, y0 = (int)fy0;
  const int x1 = x0 + 1,  y1 = y0 + 1;
  const float wx = ix - fx0, wy = iy - fy0;

  const float* fp = feat + ((size_t)(b * C_ + c)) * (size_t)(H_ * W_);
  auto corner = [&](int yy, int xx) -> float {
    const bool v = (xx >= 0) & (xx < W_) & (yy >= 0) & (yy < H_);
    const int yc = min(max(yy, 0), H_ - 1);
    const int xc = min(max(xx, 0), W_ - 1);
    const float d = fp[yc * W_ + xc];
    return v ? d : 0.0f;
  };
  const float v00 = corner(y0, x0), v01 = corner(y0, x1);
  const float v10 = corner(y1, x0), v11 = corner(y1, x1);
  const float samp = v00 * (1.0f - wx) * (1.0f - wy) + v01 * wx * (1.0f - wy)
                   + v10 * (1.0f - wx) * wy + v11 * wx * wy;

  float acc = 0.0f;
#pragma unroll
  for (int i = 0; i < 9; ++i) acc = fmaf(wv[i], samp, acc);
  const float val = acc / wv[9];
  mid[((size_t)((b * OH_ + oy) * OW_) + ox) * C_ + c] = val;
}

// ---------------------------------------------------------------------------
// TDM: DMA a [288 rows x 32 f32] weight tile (row stride `row_elems`) from
// global into LDS at byte offset `lds_off`. D# per CDNA5 ISA ch.8 (2D tensor,
// count=1, type=2, data_size=4B). Tracked with TENSORcnt. 6-arg builtin:
//   (uint32x4 g0, int32x8 g1, int32x4 g2, int32x4 g3, int32x8 g4, i32 cpol)
// ---------------------------------------------------------------------------
__device__ __forceinline__ void tdm_load_weights(const float* gsrc, unsigned lds_off,
                                                 int row_elems) {
  const uint64_t ga = (uint64_t)(uintptr_t)gsrc;
  v4u g0;
  g0.x = 1u;                                                // count=1, user mode
  g0.y = lds_off;                                           // LDS byte address of dest
  g0.z = (unsigned)(ga & 0xffffffffu);                      // global_addr[31:0]
  g0.w = (unsigned)((ga >> 32) & 0x1ffffffu) | (2u << 30);  // global_addr[56:32], type=2
  const unsigned dim0 = (unsigned)row_elems;                // tensor_dim0 (elements)
  const unsigned dim1 = (unsigned)KPAIRS;                   // tensor_dim1 = 288 rows
  const unsigned tile0 = 32u, tile1 = (unsigned)KPAIRS;
  v8i g1;
  g1[0] = (int)(2u << 16);                                  // data_size = 4 bytes
  g1[1] = (int)((dim0 & 0xffffu) << 16);                    // tensor_dim0[15:0]
  g1[2] = (int)(((dim0 >> 16) & 0xffffu) | ((dim1 & 0xffffu) << 16));
  g1[3] = (int)(((dim1 >> 16) & 0xffffu) | (tile0 << 16));  // tile_dim0 = 32
  g1[4] = (int)(tile1 & 0xffffu);                           // tile_dim1 = 288, tile_dim2 = 0
  g1[5] = (int)dim0;                                        // tensor_dim0_stride (elements)
  g1[6] = 0;
  g1[7] = 0;
  const v4i gz4 = {0, 0, 0, 0};
  const v8i gz8 = {0, 0, 0, 0, 0, 0, 0, 0};
  __builtin_amdgcn_tensor_load_to_lds(g0, g1, gz4, gz4, gz8, 0);
}

// ---------------------------------------------------------------------------
// 3x3 conv as implicit GEMM with V_WMMA_F32_16X16X4_F32.
// M = 16 consecutive pixels in one row, N = 16 output channels,
// K = 576 (c innermost -> contiguous float2 A loads from NHWC input).
// A lane layout per ISA: lanes 0-15 carry K+0/K+1, lanes 16-31 K+2/K+3.
// ---------------------------------------------------------------------------
__global__ void conv3x3_wmma_kernel(const float* __restrict__ in,   // NHWC [B][128][128][64]
                                    const float* __restrict__ bp,   // packed weights
                                    const float* __restrict__ bias,
                                    float* __restrict__ out,
                                    int row_elems,                   // 128 (conv1) / 32 (conv2)
                                    int final_nchw) {                // 1 => NCHW 3-channel out
  __shared__ float Bs[KPAIRS * 32];                                  // 36864 B

  const int nt = blockIdx.y;                                         // output-channel tile
  // Low 32 bits of a generic (flat) LDS address are the LDS byte offset.
  // ptrtoint also escapes Bs so its DMA-written contents can't be folded away.
  const unsigned lds_off = (unsigned)(uintptr_t)(void*)&Bs[0];
  if (threadIdx.x < 32) {
    tdm_load_weights(bp + nt * 32, lds_off, row_elems);
    __builtin_amdgcn_s_wait_tensorcnt(0);
  }
  __syncthreads();
  asm volatile("" ::: "memory");   // Bs now holds DMA'd weights

  const int wave = threadIdx.x >> 5;
  const int lane = threadIdx.x & 31;
  const int m    = lane & 15;    // pixel within tile (A rows)
  const int kp   = lane >> 4;    // K half: 0 -> K+0/1, 1 -> K+2/3
  const int n    = lane & 15;    // output channel within tile (B/D cols)

  const int t   = blockIdx.x * 8 + wave;   // pixel tile id, 0..8191
  const int b   = t >> 10;
  const int rem = t & 1023;
  const int y   = rem >> 3;
  const int x0  = (rem & 7) << 4;
  const int x   = x0 + m;

  const v2f zero2 = {0.0f, 0.0f};
  v8f acc = {};
#pragma unroll
  for (int ky = 0; ky < 3; ++ky) {
    const int yy = y + ky - 1;
    const bool rv = (yy >= 0) && (yy < OH_);
    const float* rowp = in + ((size_t)(b * OH_ + (rv ? yy : 0)) * OW_) * C_;
#pragma unroll
    for (int kx = 0; kx < 3; ++kx) {
      const int xs = x + kx - 1;
      const bool valid = rv && (xs >= 0) && (xs < OW_);
      // clamped (always in-bounds) address; zero-padding applied to the value
      const float* ap = rowp + (size_t)(valid ? xs : 0) * C_ + kp * 2;
      const int s = ky * 3 + kx;
#pragma unroll
      for (int c = 0; c < C_; c += 4) {
        v2f a = *(const v2f*)(ap + c);            // unconditional load
        a = valid ? a : zero2;                    // v_cndmask, EXEC stays full
        const int kkbase = (s * C_ + c) >> 1;     // K-pair index
        const v2f bb = *(const v2f*)(&Bs[(kkbase + kp) * 32 + n * 2]);
        acc = __builtin_amdgcn_wmma_f32_16x16x4_f32(
            /*neg_a=*/false, a, /*neg_b=*/false, bb,
            /*c_mod=*/(short)0, acc, /*reuse_a=*/false, /*reuse_b=*/false);
      }
    }
  }

  // D layout: lanes 0-15 -> M = r, lanes 16-31 -> M = 8 + r, N = lane & 15.
  const int mbase = (lane >> 4) * 8;
  if (!final_nchw) {
    const float bval = bias[nt * 16 + n];
#pragma unroll
    for (int r = 0; r < 8; ++r) {
      const int M = mbase + r;
      out[((size_t)((b * OH_ + y) * OW_) + x0 + M) * C_ + nt * 16 + n] = acc[r] + bval;
    }
  } else if (n < 3) {
    const float bval = bias[n];
#pragma unroll
    for (int r = 0; r < 8; ++r) {
      const int M = mbase + r;
      out[(((size_t)(b * 3 + n) * OH_) + y) * OW_ + x0 + M] = acc[r] + bval;
    }
  }
}

// ---------------------------------------------------------------------------
extern "C" void kernel_launch(void* const* d_in, const int* in_sizes, int n_in,
                              void* d_out, int out_size, void* d_ws, size_t ws_size,
                              hipStream_t stream) {
  (void)in_sizes; (void)n_in; (void)out_size; (void)ws_size;
  const float* feat   = (const float*)d_in[0];
  const float* w1     = (const float*)d_in[1];
  const float* b1     = (const float*)d_in[2];
  const float* w2     = (const float*)d_in[3];
  const float* b2     = (const float*)d_in[4];
  const float* w3     = (const float*)d_in[5];
  const float* b3     = (const float*)d_in[6];
  const float* conv1w = (const float*)d_in[7];
  const float* conv1b = (const float*)d_in[8];
  const float* conv2w = (const float*)d_in[9];
  const float* conv2b = (const float*)d_in[10];

  char* ws   = (char*)d_ws;
  float* wv   = (float*)(ws);                                 // 10 floats
  float* bp1  = (float*)(ws + 256);                           // 147456 B
  float* bp2  = (float*)(ws + 256 + 147456);                  // 36864 B
  float* mid  = (float*)(ws + 184576);                        // 33554432 B (NHWC resample)
  float* mid2 = (float*)(ws + 184576 + 33554432);             // 33554432 B (NHWC conv1 out)

  mlp_weights_kernel<<<1, 256, 0, stream>>>(w1, b1, w2, b2, w3, b3, wv);
  pack_weights_kernel<<<180, 256, 0, stream>>>(conv1w, conv2w, bp1, bp2);
  resample_kernel<<<32768, 256, 0, stream>>>(feat, wv, mid);
  conv3x3_wmma_kernel<<<dim3(1024, 4), 256, 0, stream>>>(mid,  bp1, conv1b, mid2,          128, 0);
  conv3x3_wmma_kernel<<<dim3(1024, 1), 256, 0, stream>>>(mid2, bp2, conv2b, (float*)d_out,  32, 1);
}